// MoleculeGNN_34376918237551
// MI455X (gfx1250) — compile-verified
//
#include <hip/hip_runtime.h>
#include <hip/hip_bf16.h>
#include <math.h>

#define NUM_GRAPHS 256

typedef float v2f __attribute__((ext_vector_type(2)));
typedef float v8f __attribute__((ext_vector_type(8)));

// ---------------------------------------------------------------------------
// utility
// ---------------------------------------------------------------------------
__global__ void k_fill(float* __restrict__ p, float val, size_t n) {
    size_t i = (size_t)blockIdx.x * blockDim.x + threadIdx.x;
    if (i < n) p[i] = val;
}

// deg[dst] += 1 for every edge (deg buffer pre-filled with 1.0 = self loop)
__global__ void k_deg_accum(const long long* __restrict__ dst,
                            float* __restrict__ deg, int E) {
    int e = blockIdx.x * blockDim.x + threadIdx.x;
    if (e < E) unsafeAtomicAdd(&deg[(int)dst[e]], 1.0f);
}

__global__ void k_rsqrt_inplace(float* __restrict__ p, int n) {
    int i = blockIdx.x * blockDim.x + threadIdx.x;
    if (i < n) p[i] = rsqrtf(p[i]);
}

// ---------------------------------------------------------------------------
// layer 1 GEMM: [N,5] @ [5,64]  (K=5 too small for WMMA; plain VALU)
// ---------------------------------------------------------------------------
__global__ void k_gemm1(const float* __restrict__ x, const float* __restrict__ W1,
                        float* __restrict__ Hpre, int N) {
    long long t = (long long)blockIdx.x * blockDim.x + threadIdx.x;
    if (t >= (long long)N * 64) return;
    int i = (int)(t >> 6);
    int j = (int)(t & 63);
    const float* xr = x + (size_t)i * 5;
    float s = 0.0f;
#pragma unroll
    for (int k = 0; k < 5; ++k) s = fmaf(xr[k], W1[k * 64 + j], s);
    Hpre[t] = s;
}

// ---------------------------------------------------------------------------
// WMMA fp32 GEMM: [N,K] @ [K,DOUT] -> [N,DOUT]
// one wave per 16x16 output tile; block = (DOUT/16) waves covering 16 rows.
// V_WMMA_F32_16X16X4_F32 per ISA 7.12.2:
//   A 16x4 : lanes 0-15 rows, VGPR0 = {K0 | K2}, VGPR1 = {K1 | K3}
//   B 4x16 : lanes 0-15 cols, VGPR0 = {K0 | K2}, VGPR1 = {K1 | K3}
//   C/D    : VGPR v -> row v (lanes 0-15) / row v+8 (lanes 16-31)
// ---------------------------------------------------------------------------
template <int K, int DOUT>
__global__ void k_gemm_wmma(const float* __restrict__ Hin,
                            const float* __restrict__ W,
                            float* __restrict__ Hpre, int N) {
    const int lane = threadIdx.x & 31;
    const int wave = threadIdx.x >> 5;
    const int rowBase = blockIdx.x * 16;
    const int colBase = wave * 16;
    const int lr = lane & 15;         // row (A) / col (B) within tile
    const int hk = (lane >> 4) * 2;   // K sub-offset: 0 for lanes 0-15, 2 for 16-31

    int row = rowBase + lr;
    if (row >= N) row = N - 1;        // clamp (loads only; keeps EXEC all-ones)
    const float* arow = Hin + (size_t)row * K;
    const int col = colBase + lr;

    v8f c = {0.f, 0.f, 0.f, 0.f, 0.f, 0.f, 0.f, 0.f};
#pragma unroll
    for (int kk = 0; kk < K; kk += 4) {
        v2f a, b;
        a.x = arow[kk + hk];
        a.y = arow[kk + hk + 1];
        b.x = W[(size_t)(kk + hk) * DOUT + col];
        b.y = W[(size_t)(kk + hk + 1) * DOUT + col];
        c = __builtin_amdgcn_wmma_f32_16x16x4_f32(
                /*neg_a=*/false, a, /*neg_b=*/false, b,
                /*c_mod=*/(short)0, c, /*reuse_a=*/false, /*reuse_b=*/false);
    }

    const int srow = rowBase + (lane >> 4) * 8;
#pragma unroll
    for (int v = 0; v < 8; ++v) {
        int r = srow + v;
        if (r < N) Hpre[(size_t)r * DOUT + col] = c[v];
    }
}

// ---------------------------------------------------------------------------
// edge scatter: Agg[dst] += Hpre[src] * dinv[src]*dinv[dst]
// thread = (edge, 4-feature chunk);  chunk = blockIdx.y
// ---------------------------------------------------------------------------
template <int D>
__global__ void k_scatter(const long long* __restrict__ srcI,
                          const long long* __restrict__ dstI,
                          const float* __restrict__ dinv,
                          const float* __restrict__ Hpre,
                          float* __restrict__ Agg, int E) {
    int e = blockIdx.x * blockDim.x + threadIdx.x;
    if (e >= E) return;
    int s = (int)srcI[e];
    int d = (int)dstI[e];
    float nrm = dinv[s] * dinv[d];
    int j = blockIdx.y * 4;
    const float4 h = *(const float4*)(Hpre + (size_t)s * D + j);
    float* out = Agg + (size_t)d * D + j;
    unsafeAtomicAdd(out + 0, h.x * nrm);
    unsafeAtomicAdd(out + 1, h.y * nrm);
    unsafeAtomicAdd(out + 2, h.z * nrm);
    unsafeAtomicAdd(out + 3, h.w * nrm);
}

// ---------------------------------------------------------------------------
// finalize: Hout = relu(Agg + Hpre*dinv^2 (self loop) + bias)
// ---------------------------------------------------------------------------
template <int D>
__global__ void k_finalize(const float* __restrict__ Agg,
                           const float* __restrict__ Hpre,
                           const float* __restrict__ dinv,
                           const float* __restrict__ bias,
                           float* __restrict__ Hout, int N) {
    long long t = (long long)blockIdx.x * blockDim.x + threadIdx.x;
    if (t >= (long long)N * D) return;
    int i = (int)(t / D);
    int j = (int)(t % D);
    float di = dinv[i];
    float v = Agg[t] + Hpre[t] * di * di + bias[j];
    Hout[t] = fmaxf(v, 0.0f);
}

// ---------------------------------------------------------------------------
// pooling: per-graph feature sums + node counts (h width = 64)
// ---------------------------------------------------------------------------
__global__ void k_pool(const long long* __restrict__ batch,
                       const float* __restrict__ h,
                       float* __restrict__ pool, float* __restrict__ cnt, int N) {
    int i = blockIdx.x * blockDim.x + threadIdx.x;
    if (i >= N) return;
    int g = (int)batch[i];
    int j = blockIdx.y * 4;
    const float4 v = *(const float4*)(h + (size_t)i * 64 + j);
    float* out = pool + (size_t)g * 64 + j;
    unsafeAtomicAdd(out + 0, v.x);
    unsafeAtomicAdd(out + 1, v.y);
    unsafeAtomicAdd(out + 2, v.z);
    unsafeAtomicAdd(out + 3, v.w);
    if (blockIdx.y == 0) unsafeAtomicAdd(&cnt[g], 1.0f);
}

// ---------------------------------------------------------------------------
// FC + sigmoid: out[g] = sigmoid(dot(pool[g]/cnt[g], Wfc) + bfc)
// ---------------------------------------------------------------------------
__global__ void k_fc(const float* __restrict__ pool, const float* __restrict__ cnt,
                     const float* __restrict__ Wfc, const float* __restrict__ bfc,
                     float* __restrict__ out) {
    int g = threadIdx.x;
    if (g >= NUM_GRAPHS) return;
    float acc = 0.0f;
#pragma unroll 8
    for (int j = 0; j < 64; ++j) acc = fmaf(pool[g * 64 + j], Wfc[j], acc);
    float c = fmaxf(cnt[g], 1.0f);
    float v = acc / c + bfc[0];
    out[g] = 1.0f / (1.0f + expf(-v));
}

// ---------------------------------------------------------------------------
// host
// ---------------------------------------------------------------------------
extern "C" void kernel_launch(void* const* d_in, const int* in_sizes, int n_in,
                              void* d_out, int out_size, void* d_ws, size_t ws_size,
                              hipStream_t stream) {
    const float* x        = (const float*)d_in[0];
    const long long* edge = (const long long*)d_in[1];
    const long long* bat  = (const long long*)d_in[2];
    const float* W1 = (const float*)d_in[3];
    const float* b1 = (const float*)d_in[4];
    const float* W2 = (const float*)d_in[5];
    const float* b2 = (const float*)d_in[6];
    const float* W3 = (const float*)d_in[7];
    const float* b3 = (const float*)d_in[8];
    const float* Wfc = (const float*)d_in[9];
    const float* bfc = (const float*)d_in[10];

    const int N = in_sizes[0] / 5;
    const int E = in_sizes[1] / 2;
    const long long* srcI = edge;       // edge_index[0]
    const long long* dstI = edge + E;   // edge_index[1]

    // workspace layout (floats), 256-float aligned slabs
    float* f = (float*)d_ws;
    size_t Np = ((size_t)N + 255) & ~(size_t)255;
    float* dinv = f;
    float* bufA = dinv + Np;             // N x 128 (transformed features Hpre)
    float* bufB = bufA + Np * 128;       // N x 128 (aggregation target)
    float* bufC = bufB + Np * 128;       // N x 128 (layer output / next input)
    float* pool = bufC + Np * 128;       // 256 x 64
    float* cnt  = pool + NUM_GRAPHS * 64;

    const int T = 256;
    auto cdiv = [](long long a, long long b) { return (int)((a + b - 1) / b); };

    // --- symmetric normalization: dinv = rsqrt(in_deg + 1) -----------------
    k_fill<<<cdiv(N, T), T, 0, stream>>>(dinv, 1.0f, (size_t)N);
    k_deg_accum<<<cdiv(E, T), T, 0, stream>>>(dstI, dinv, E);
    k_rsqrt_inplace<<<cdiv(N, T), T, 0, stream>>>(dinv, N);

    // --- layer 1: x[N,5]@W1[5,64] -> A; scatter -> B; relu -> C ------------
    k_gemm1<<<cdiv((long long)N * 64, T), T, 0, stream>>>(x, W1, bufA, N);
    k_fill<<<cdiv((long long)N * 64, T), T, 0, stream>>>(bufB, 0.0f, (size_t)N * 64);
    {   dim3 g(cdiv(E, T), 64 / 4);
        k_scatter<64><<<g, T, 0, stream>>>(srcI, dstI, dinv, bufA, bufB, E); }
    k_finalize<64><<<cdiv((long long)N * 64, T), T, 0, stream>>>(bufB, bufA, dinv, b1, bufC, N);

    // --- layer 2: C[N,64]@W2[64,128] (WMMA) -> A; scatter -> B; relu -> C --
    k_gemm_wmma<64, 128><<<cdiv(N, 16), 32 * (128 / 16), 0, stream>>>(bufC, W2, bufA, N);
    k_fill<<<cdiv((long long)N * 128, T), T, 0, stream>>>(bufB, 0.0f, (size_t)N * 128);
    {   dim3 g(cdiv(E, T), 128 / 4);
        k_scatter<128><<<g, T, 0, stream>>>(srcI, dstI, dinv, bufA, bufB, E); }
    k_finalize<128><<<cdiv((long long)N * 128, T), T, 0, stream>>>(bufB, bufA, dinv, b2, bufC, N);

    // --- layer 3: C[N,128]@W3[128,64] (WMMA) -> A; scatter -> B; relu -> C -
    k_gemm_wmma<128, 64><<<cdiv(N, 16), 32 * (64 / 16), 0, stream>>>(bufC, W3, bufA, N);
    k_fill<<<cdiv((long long)N * 64, T), T, 0, stream>>>(bufB, 0.0f, (size_t)N * 64);
    {   dim3 g(cdiv(E, T), 64 / 4);
        k_scatter<64><<<g, T, 0, stream>>>(srcI, dstI, dinv, bufA, bufB, E); }
    k_finalize<64><<<cdiv((long long)N * 64, T), T, 0, stream>>>(bufB, bufA, dinv, b3, bufC, N);

    // --- global mean pool + FC + sigmoid -----------------------------------
    k_fill<<<cdiv(NUM_GRAPHS * 64 + NUM_GRAPHS, T), T, 0, stream>>>(
        pool, 0.0f, (size_t)NUM_GRAPHS * 64 + NUM_GRAPHS);
    {   dim3 g(cdiv(N, T), 64 / 4);
        k_pool<<<g, T, 0, stream>>>(bat, bufC, pool, cnt, N); }
    k_fc<<<1, NUM_GRAPHS, 0, stream>>>(pool, cnt, Wfc, bfc, (float*)d_out);
}